// GATLayer_34574486733588
// MI455X (gfx1250) — compile-verified
//
#include <hip/hip_runtime.h>
#include <math.h>

typedef __attribute__((ext_vector_type(2))) float v2f;
typedef __attribute__((ext_vector_type(8))) float v8f;

#define HC 128
#define NHEAD 4
#define CH 32
#define EPS 1e-5f
#define NEG_SLOPE 0.2f

// ---------- monotone float <-> uint key (for atomic float max) ----------
__device__ __forceinline__ unsigned fkey(float f) {
    unsigned u = __float_as_uint(f);
    return (u & 0x80000000u) ? ~u : (u | 0x80000000u);
}
__device__ __forceinline__ float kinv(unsigned k) {
    return (k & 0x80000000u) ? __uint_as_float(k ^ 0x80000000u)
                             : __uint_as_float(~k);
}
__device__ __forceinline__ float lrelu(float a) {
    return a >= 0.f ? a : NEG_SLOPE * a;
}

// ---------- fp32 WMMA GEMM: C[N x 128] = A[N x 128] * B[128 x 128] (+bias) ----
// block = 256 threads = 8 waves; each wave owns one 16-wide column tile.
// grid.x = ceil(N/16) row tiles.
__global__ void gemm128_wmma(const float* __restrict__ A,
                             const float* __restrict__ B,
                             const float* __restrict__ bias,
                             float* __restrict__ C, int N) {
    const int lane = threadIdx.x & 31;
    const int wv   = threadIdx.x >> 5;      // 0..7 -> column tile
    const int m0   = blockIdx.x * 16;
    const int n0   = wv * 16;
    const int hi   = lane >> 4;             // 0 or 1 (lane half)
    const int l16  = lane & 15;

    int row = m0 + l16;
    if (row >= N) row = N - 1;              // clamp, keeps EXEC uniform
    const float* arow = A + (size_t)row * HC;

    v8f acc = {};
    for (int k = 0; k < HC; k += 4) {
        v2f a, b;
        // A 16x4: VGPR0 = K = k+2*hi, VGPR1 = K = k+2*hi+1 (per-lane-half)
        a.x = arow[k + 2 * hi];
        a.y = arow[k + 2 * hi + 1];
        // B 4x16: symmetric striping over N within each lane half
        b.x = B[(size_t)(k + 2 * hi) * HC + n0 + l16];
        b.y = B[(size_t)(k + 2 * hi + 1) * HC + n0 + l16];
        acc = __builtin_amdgcn_wmma_f32_16x16x4_f32(
            false, a, false, b, (short)0, acc, false, false);
    }
    // D layout: lanes 0-15 -> M = r, lanes 16-31 -> M = r+8
    const float bv = bias ? bias[n0 + l16] : 0.f;
    float* cptr = C + (size_t)(m0 + 8 * hi) * HC + n0 + l16;
    if (m0 + 16 <= N) {               // uniform fast path: full row tile
#pragma unroll
        for (int r = 0; r < 8; ++r)
            cptr[(size_t)r * HC] = acc[r] + bv;
    } else {                          // tail tile (not hit for N % 16 == 0)
        for (int r = 0; r < 8; ++r)
            if (m0 + 8 * hi + r < N)
                cptr[(size_t)r * HC] = acc[r] + bv;
    }
}

// ---------- per-(node,head) attention logits ----------
__global__ void attn_node(const float* __restrict__ h,
                          const float* __restrict__ att_src,
                          const float* __restrict__ att_dst,
                          float* __restrict__ a_src,
                          float* __restrict__ a_dst, int N) {
    int idx = blockIdx.x * blockDim.x + threadIdx.x;
    if (idx >= N * NHEAD) return;
    int n = idx >> 2, hh = idx & 3;
    const float* hp = h + (size_t)n * HC + hh * CH;
    const float* as = att_src + hh * CH;
    const float* ad = att_dst + hh * CH;
    float s = 0.f, d = 0.f;
#pragma unroll
    for (int c = 0; c < CH; ++c) {
        float v = hp[c];
        s += v * as[c];
        d += v * ad[c];
    }
    a_src[idx] = s;
    a_dst[idx] = d;
}

// ---------- init: out_pre = bias, max keys = 0 (== -inf), denom = 0 ----------
__global__ void init_ws(float* __restrict__ out_pre,
                        const float* __restrict__ bias,
                        unsigned* __restrict__ mkey,
                        float* __restrict__ denom, int N) {
    int idx = blockIdx.x * blockDim.x + threadIdx.x;
    if (idx < N * HC) out_pre[idx] = bias[idx & (HC - 1)];
    if (idx < N * NHEAD) { mkey[idx] = 0u; denom[idx] = 0.f; }
}

// ---------- segment max over dst (edges + implicit self-loops) ----------
__global__ void edge_max(const int* __restrict__ ei,
                         const float4* __restrict__ a_src4,
                         const float4* __restrict__ a_dst4,
                         unsigned* __restrict__ mkey, int E, int N) {
    int e = blockIdx.x * blockDim.x + threadIdx.x;
    int Et = E + N;
    if (e >= Et) return;
    int s, d;
    if (e < E) { s = ei[e]; d = ei[E + e]; } else { s = d = e - E; }
    float4 as = a_src4[s];
    float4 ad = a_dst4[d];
    unsigned* mp = &mkey[(size_t)d * NHEAD];
    atomicMax(&mp[0], fkey(lrelu(as.x + ad.x)));
    atomicMax(&mp[1], fkey(lrelu(as.y + ad.y)));
    atomicMax(&mp[2], fkey(lrelu(as.z + ad.z)));
    atomicMax(&mp[3], fkey(lrelu(as.w + ad.w)));
}

// ---------- segment sum of exp(alpha - m) ----------
__global__ void edge_den(const int* __restrict__ ei,
                         const float4* __restrict__ a_src4,
                         const float4* __restrict__ a_dst4,
                         const uint4* __restrict__ mkey4,
                         float* __restrict__ denom, int E, int N) {
    int e = blockIdx.x * blockDim.x + threadIdx.x;
    int Et = E + N;
    if (e >= Et) return;
    int s, d;
    if (e < E) { s = ei[e]; d = ei[E + e]; } else { s = d = e - E; }
    float4 as = a_src4[s];
    float4 ad = a_dst4[d];
    uint4  mk = mkey4[d];
    float* dp = &denom[(size_t)d * NHEAD];
    atomicAdd(&dp[0], __expf(lrelu(as.x + ad.x) - kinv(mk.x)));
    atomicAdd(&dp[1], __expf(lrelu(as.y + ad.y) - kinv(mk.y)));
    atomicAdd(&dp[2], __expf(lrelu(as.z + ad.z) - kinv(mk.z)));
    atomicAdd(&dp[3], __expf(lrelu(as.w + ad.w) - kinv(mk.w)));
}

// ---------- message scatter: wave per edge, lane = channel within head ----------
__global__ void edge_scatter(const int* __restrict__ ei,
                             const float4* __restrict__ a_src4,
                             const float4* __restrict__ a_dst4,
                             const uint4* __restrict__ mkey4,
                             const float4* __restrict__ denom4,
                             const float* __restrict__ h,
                             float* __restrict__ out_pre, int E, int N) {
    int lane = threadIdx.x & 31;
    int e = blockIdx.x * (blockDim.x >> 5) + (threadIdx.x >> 5);
    int Et = E + N;
    if (e >= Et) return;
    int s, d;
    if (e < E) { s = ei[e]; d = ei[E + e]; } else { s = d = e - E; }
    float4 as = a_src4[s];
    float4 ad = a_dst4[d];
    uint4  mk = mkey4[d];
    float4 dn = denom4[d];
    float alpha[NHEAD];
    alpha[0] = __expf(lrelu(as.x + ad.x) - kinv(mk.x)) / dn.x;
    alpha[1] = __expf(lrelu(as.y + ad.y) - kinv(mk.y)) / dn.y;
    alpha[2] = __expf(lrelu(as.z + ad.z) - kinv(mk.z)) / dn.z;
    alpha[3] = __expf(lrelu(as.w + ad.w) - kinv(mk.w)) / dn.w;
    const float* hs = h + (size_t)s * HC;
    float* op = out_pre + (size_t)d * HC;
#pragma unroll
    for (int hh = 0; hh < NHEAD; ++hh) {
        int j = hh * CH + lane;
        atomicAdd(&op[j], alpha[hh] * hs[j]);
    }
}

// ---------- BatchNorm column stats for both tensors ----------
// grid.x = 256: b<128 -> out_pre col b ; b>=128 -> ident col (b-128)
__global__ void bn_stats(const float* __restrict__ out_pre,
                         const float* __restrict__ ident,
                         float* __restrict__ stats, int N) {
    __shared__ float ssum[256], ssq[256];
    int b = blockIdx.x;
    const float* src = (b < HC) ? out_pre : ident;
    int col = b & (HC - 1);
    float s = 0.f, q = 0.f;
    for (int n = threadIdx.x; n < N; n += blockDim.x) {
        float v = src[(size_t)n * HC + col];
        s += v; q += v * v;
    }
    ssum[threadIdx.x] = s; ssq[threadIdx.x] = q;
    __syncthreads();
    for (int st = 128; st > 0; st >>= 1) {
        if ((int)threadIdx.x < st) {
            ssum[threadIdx.x] += ssum[threadIdx.x + st];
            ssq[threadIdx.x]  += ssq[threadIdx.x + st];
        }
        __syncthreads();
    }
    if (threadIdx.x == 0) {
        float mu  = ssum[0] / (float)N;
        float var = ssq[0] / (float)N - mu * mu;
        stats[b * 2]     = mu;
        stats[b * 2 + 1] = rsqrtf(var + EPS);
    }
}

// ---------- fused BN -> LN -> residual-BN -> ReLU, one wave per row ----------
__global__ void finalize(const float* __restrict__ out_pre,
                         const float* __restrict__ ident,
                         const float* __restrict__ stats,
                         const float* __restrict__ bn_g, const float* __restrict__ bn_b,
                         const float* __restrict__ ln_g, const float* __restrict__ ln_b,
                         const float* __restrict__ bnr_g, const float* __restrict__ bnr_b,
                         float* __restrict__ out, int N) {
    int lane = threadIdx.x & 31;
    int n = blockIdx.x * (blockDim.x >> 5) + (threadIdx.x >> 5);
    if (n >= N) return;
    float t[4];
    float s = 0.f, q = 0.f;
#pragma unroll
    for (int k = 0; k < 4; ++k) {
        int j = lane + 32 * k;
        float v = out_pre[(size_t)n * HC + j];
        float tv = (v - stats[j * 2]) * stats[j * 2 + 1] * bn_g[j] + bn_b[j];
        t[k] = tv; s += tv; q += tv * tv;
    }
#pragma unroll
    for (int m = 16; m > 0; m >>= 1) {
        s += __shfl_xor(s, m, 32);
        q += __shfl_xor(q, m, 32);
    }
    float mu  = s * (1.f / HC);
    float var = q * (1.f / HC) - mu * mu;
    float r   = rsqrtf(var + EPS);
#pragma unroll
    for (int k = 0; k < 4; ++k) {
        int j = lane + 32 * k;
        float u = (t[k] - mu) * r * ln_g[j] + ln_b[j];
        float iv = ident[(size_t)n * HC + j];
        float w = (iv - stats[2 * HC + j * 2]) * stats[2 * HC + j * 2 + 1] * bnr_g[j] + bnr_b[j];
        float z = u + w;
        out[(size_t)n * HC + j] = z > 0.f ? z : 0.f;
    }
}

extern "C" void kernel_launch(void* const* d_in, const int* in_sizes, int n_in,
                              void* d_out, int out_size, void* d_ws, size_t ws_size,
                              hipStream_t stream) {
    const float* x       = (const float*)d_in[0];
    const int*   ei      = (const int*)d_in[1];
    const float* W       = (const float*)d_in[2];
    const float* att_src = (const float*)d_in[3];
    const float* att_dst = (const float*)d_in[4];
    const float* bias    = (const float*)d_in[5];
    const float* bn_g    = (const float*)d_in[6];
    const float* bn_b    = (const float*)d_in[7];
    const float* ln_g    = (const float*)d_in[8];
    const float* ln_b    = (const float*)d_in[9];
    const float* res_W   = (const float*)d_in[10];
    const float* res_b   = (const float*)d_in[11];
    const float* bnr_g   = (const float*)d_in[12];
    const float* bnr_b   = (const float*)d_in[13];
    float* out = (float*)d_out;

    const int N = in_sizes[0] / HC;   // 50000
    const int E = in_sizes[1] / 2;    // 800000
    const int Et = E + N;

    // workspace layout (all chunks 16B-aligned)
    size_t fN = (size_t)N * HC;
    float* h      = (float*)d_ws;
    float* ident  = h + fN;
    float* outp   = ident + fN;
    float* a_src  = outp + fN;
    float* a_dst  = a_src + (size_t)N * NHEAD;
    float* denom  = a_dst + (size_t)N * NHEAD;
    float* stats  = denom + (size_t)N * NHEAD;
    unsigned* mkey = (unsigned*)(stats + 4 * HC);

    const float4* a_src4 = (const float4*)a_src;
    const float4* a_dst4 = (const float4*)a_dst;
    const uint4*  mkey4  = (const uint4*)mkey;
    const float4* denom4 = (const float4*)denom;

    dim3 blk(256);

    // 1) dense projections via fp32 WMMA
    gemm128_wmma<<<(N + 15) / 16, blk, 0, stream>>>(x, W, nullptr, h, N);
    gemm128_wmma<<<(N + 15) / 16, blk, 0, stream>>>(x, res_W, res_b, ident, N);

    // 2) per-node attention logits
    attn_node<<<(N * NHEAD + 255) / 256, blk, 0, stream>>>(h, att_src, att_dst, a_src, a_dst, N);

    // 3) init accumulators
    init_ws<<<((int)fN + 255) / 256, blk, 0, stream>>>(outp, bias, mkey, denom, N);

    // 4) segment max
    edge_max<<<(Et + 255) / 256, blk, 0, stream>>>(ei, a_src4, a_dst4, mkey, E, N);

    // 5) segment softmax denominator
    edge_den<<<(Et + 255) / 256, blk, 0, stream>>>(ei, a_src4, a_dst4, mkey4, denom, E, N);

    // 6) weighted message scatter (wave per edge)
    edge_scatter<<<(Et + 7) / 8, blk, 0, stream>>>(ei, a_src4, a_dst4, mkey4, denom4, h, outp, E, N);

    // 7) BN stats (both tensors)
    bn_stats<<<2 * HC, blk, 0, stream>>>(outp, ident, stats, N);

    // 8) fused normalize + residual + relu
    finalize<<<(N + 7) / 8, blk, 0, stream>>>(outp, ident, stats,
                                              bn_g, bn_b, ln_g, ln_b, bnr_g, bnr_b, out, N);
}